// DescriptorModule_81827716923470
// MI455X (gfx1250) — compile-verified
//
#include <hip/hip_runtime.h>
#include <hip/hip_bf16.h>

typedef __attribute__((ext_vector_type(2))) float v2f;
typedef __attribute__((ext_vector_type(8))) float v8f;

#define NUM_TYPES 4
#define NDESC 384
#define ATOMS_PER_CONF 64

// ---------------------------------------------------------------------------
// Branch-free tanh. gfx1250 has V_TANH_F32; fall back to a divergence-free
// exp2-based formulation if the builtin is not declared by this toolchain.
// ---------------------------------------------------------------------------
__device__ __forceinline__ float fast_tanh(float x) {
#if __has_builtin(__builtin_amdgcn_tanhf)
    return __builtin_amdgcn_tanhf(x);
#elif __has_builtin(__builtin_amdgcn_tanh_f32)
    return __builtin_amdgcn_tanh_f32(x);
#else
    // tanh(x) = sign(x) * (1 - 2t/(1+t)), t = exp(-2|x|)  -- no branches
    float ax = fabsf(x);
    float t  = __expf(-2.0f * ax);          // v_exp_f32 based, branch-free
    float r  = 1.0f - 2.0f * t * __frcp_rn(1.0f + t);
    return copysignf(r, x);
#endif
}

// ---------------------------------------------------------------------------
// Kernel 1: per-block type histogram (256 atoms / block)
// ---------------------------------------------------------------------------
__global__ void count_kernel(const int* __restrict__ types,
                             int* __restrict__ blockCounts, int n) {
    __shared__ int cnt[NUM_TYPES];
    int t = threadIdx.x;
    if (t < NUM_TYPES) cnt[t] = 0;
    __syncthreads();
    int i = blockIdx.x * 256 + t;
    if (i < n) atomicAdd(&cnt[types[i] & (NUM_TYPES - 1)], 1);
    __syncthreads();
    if (t < NUM_TYPES) blockCounts[blockIdx.x * NUM_TYPES + t] = cnt[t];
}

// ---------------------------------------------------------------------------
// Kernel 2: single-block scan: per-block running offsets per type, type bases,
// and the float-typed per-type counts in the output tail.
// ---------------------------------------------------------------------------
__global__ void scan_kernel(const int* __restrict__ blockCounts,
                            int* __restrict__ blockOffsets,
                            int* __restrict__ typeBase,
                            int* __restrict__ typeCount,
                            float* __restrict__ outCounts, int numBlocks) {
    int t = threadIdx.x;
    if (t < NUM_TYPES) {
        int run = 0;
        for (int b = 0; b < numBlocks; ++b) {
            blockOffsets[b * NUM_TYPES + t] = run;
            run += blockCounts[b * NUM_TYPES + t];
        }
        typeCount[t] = run;
    }
    __syncthreads();
    if (t == 0) {
        int base = 0;
        for (int q = 0; q < NUM_TYPES; ++q) { typeBase[q] = base; base += typeCount[q]; }
    }
    __syncthreads();
    if (t < NUM_TYPES) outCounts[t] = (float)typeCount[t];
}

// ---------------------------------------------------------------------------
// Kernel 3: stable in-block rank -> global sorted position; scatter conformer id
// ---------------------------------------------------------------------------
__global__ void rank_kernel(const int* __restrict__ types,
                            const int* __restrict__ blockOffsets,
                            const int* __restrict__ typeBase,
                            int* __restrict__ pos,
                            float* __restrict__ outConf, int n) {
    __shared__ int sty[256];
    int t = threadIdx.x;
    int i = blockIdx.x * 256 + t;
    int ty = (i < n) ? (types[i] & (NUM_TYPES - 1)) : 0;
    sty[t] = ty;
    __syncthreads();
    if (i < n) {
        int rank = 0;
        for (int j = 0; j < t; ++j) rank += (sty[j] == ty);
        int p = typeBase[ty] + blockOffsets[blockIdx.x * NUM_TYPES + ty] + rank;
        pos[i] = p;
        __builtin_nontemporal_store((float)(i / ATOMS_PER_CONF), outConf + p);
    }
}

// ---------------------------------------------------------------------------
// Kernel 4: fused desc = tanh(coords @ W) scattered to sorted rows.
// One wave32 per 16-atom tile; V_WMMA_F32_16X16X4_F32, K padded 3 -> 4.
//
// A (16x4 f32): lanes 0-15 VGPR0=K0(x) VGPR1=K1(y), row M=lane
//               lanes 16-31 VGPR0=K2(z) VGPR1=K3(0), row M=lane-16
// B (4x16 f32): lanes 0-15 VGPR0=rowK0 VGPR1=rowK1, col N=lane
//               lanes 16-31 VGPR0=rowK2 VGPR1=rowK3(0), col N=lane-16
// D (16x16 f32): VGPR v = row M=v (+8 for lanes>=16), col N=lane%16.
//
// All lane-variant data is loaded unconditionally and combined with selects so
// EXEC stays all-1s throughout (WMMA requirement, and no saveexec overhead).
// ---------------------------------------------------------------------------
__global__ void __launch_bounds__(256)
desc_wmma_kernel(const float* __restrict__ coords,
                 const float* __restrict__ W,
                 const int* __restrict__ pos,
                 float* __restrict__ out, int n) {
    const int lane = threadIdx.x & 31;
    const int wave = blockIdx.x * (blockDim.x >> 5) + (threadIdx.x >> 5);
    const int base = wave * 16;
    if (base >= n) return;                 // uniform per wave

    const bool hi = lane >= 16;
    const int  l  = lane & 15;

    // --- A fragment: unconditional x,y,z load, then select ------------------
    int ma = base + l; if (ma >= n) ma = n - 1;
    const float* c3 = coords + (size_t)ma * 3;
    const float cx = c3[0], cy = c3[1], cz = c3[2];
    v2f a;
    a.x = hi ? cz   : cx;
    a.y = hi ? 0.0f : cy;

    // --- sorted-destination row pointers for the 8 rows this lane stores ---
    float* rp[8];
    const int rbase = base + (hi ? 8 : 0);
#pragma unroll
    for (int v = 0; v < 8; ++v) {
        int r = rbase + v; if (r >= n) r = n - 1;
        rp[v] = out + (size_t)pos[r] * NDESC;
    }

    // --- 24 WMMAs across the 384 descriptor columns -------------------------
    for (int n0 = 0; n0 < NDESC; n0 += 16) {
        const int col = n0 + l;
        // unconditional loads (valid addresses for every lane), then select
        const float wa = W[(hi ? 2 : 0) * NDESC + col];  // row K0 or K2
        const float wb = W[1 * NDESC + col];             // row K1 (always read)
        v2f b;
        b.x = wa;
        b.y = hi ? 0.0f : wb;

        v8f c = {};
        c = __builtin_amdgcn_wmma_f32_16x16x4_f32(
                /*neg_a=*/false, a, /*neg_b=*/false, b,
                /*c_mod=*/(short)0, c, /*reuse_a=*/false, /*reuse_b=*/false);

#pragma unroll
        for (int v = 0; v < 8; ++v)
            __builtin_nontemporal_store(fast_tanh(c[v]), rp[v] + col);
    }
}

// ---------------------------------------------------------------------------
extern "C" void kernel_launch(void* const* d_in, const int* in_sizes, int n_in,
                              void* d_out, int out_size, void* d_ws, size_t ws_size,
                              hipStream_t stream) {
    const float* coords = (const float*)d_in[0];   // (n_confs, 64, 3) f32
    const int*   types  = (const int*)d_in[1];     // (n_confs, 64)    int
    const float* W      = (const float*)d_in[2];   // (3, 384)         f32

    const int n = in_sizes[1];                     // total atoms = 131072
    const int numBlocks = (n + 255) / 256;

    // workspace layout (ints)
    int* pos          = (int*)d_ws;                          // n
    int* blockCounts  = pos + n;                             // numBlocks*4
    int* blockOffsets = blockCounts + numBlocks * NUM_TYPES; // numBlocks*4
    int* typeBase     = blockOffsets + numBlocks * NUM_TYPES;// 4
    int* typeCount    = typeBase + NUM_TYPES;                // 4

    // output layout (floats): desc_sorted | at2conf_sorted | at_type_count
    float* outDesc   = (float*)d_out;
    float* outConf   = outDesc + (size_t)n * NDESC;
    float* outCounts = outConf + n;

    count_kernel<<<numBlocks, 256, 0, stream>>>(types, blockCounts, n);
    scan_kernel<<<1, 64, 0, stream>>>(blockCounts, blockOffsets, typeBase,
                                      typeCount, outCounts, numBlocks);
    rank_kernel<<<numBlocks, 256, 0, stream>>>(types, blockOffsets, typeBase,
                                               pos, outConf, n);

    const int tiles       = (n + 15) / 16;        // one wave per 16-atom tile
    const int wavesPerBlk = 8;                    // 256 threads = 8 wave32
    const int descBlocks  = (tiles + wavesPerBlk - 1) / wavesPerBlk;
    desc_wmma_kernel<<<descBlocks, 256, 0, stream>>>(coords, W, pos, outDesc, n);
}